// GCNEncoder3_35201551958717
// MI455X (gfx1250) — compile-verified
//
#include <hip/hip_runtime.h>

// ---------------------------------------------------------------------------
// GCN 3-layer forward for MI455X (gfx1250, wave32).
//   h = X @ W (fp32 WMMA 16x16x4, 32x64 register-blocked per wave)
//   out = bias + selfloop + edge scatter (f32 atomics resolving in L2)
// ---------------------------------------------------------------------------

typedef float v2f __attribute__((ext_vector_type(2)));
typedef float v8f __attribute__((ext_vector_type(8)));

#define GCN_N_NODES 100000

// ---------------- degree / normalization ----------------

__global__ void deg_init_kernel(float* __restrict__ deg, int n) {
  int i = blockIdx.x * blockDim.x + threadIdx.x;
  if (i < n) deg[i] = 1.0f;  // self-loop
}

__global__ void deg_edges_kernel(const int* __restrict__ dst, float* __restrict__ deg, int E) {
  int i = blockIdx.x * blockDim.x + threadIdx.x;
  if (i < E) atomicAdd(&deg[dst[i]], 1.0f);
}

__global__ void deg_rsqrt_kernel(float* __restrict__ deg, int n) {
  int i = blockIdx.x * blockDim.x + threadIdx.x;
  if (i < n) {
    float d = deg[i];
    deg[i] = (d > 0.0f) ? rsqrtf(d) : 0.0f;
  }
}

// ---------------- dense GEMM via f32 WMMA, 32x64 per wave ----------------
// Each wave owns a 32(M) x 64(N) output block = 2x4 WMMA tiles.
// Per K-step of 4: two v2f A fragments (reused over 4 N tiles) and four
// B fragment pairs (reused over 2 M tiles) feed 8 v_wmma_f32_16x16x4_f32.
//
// A fragment (16x4): lanes 0-15 hold M=lane, {K+0,K+1};
//                    lanes 16-31 hold M=lane-16, {K+2,K+3}.  (ISA 7.12.2)
// B fragment mirrored. C/D: VGPR v -> M = v (+8 for hi lanes), N = lane&15.

__global__ void gemm_wmma_f32(const float* __restrict__ A, const float* __restrict__ B,
                              float* __restrict__ D, int M, int Ncols, int K) {
  const int lane = threadIdx.x & 31;
  const int wave = threadIdx.x >> 5;
  const int wavesPerBlock = blockDim.x >> 5;
  const long grp = (long)blockIdx.x * wavesPerBlock + wave;
  const int ngrpN = Ncols >> 6;                    // 64-wide N groups
  const long totalGrp = (long)(M >> 5) * ngrpN;    // 32-tall M groups
  if (grp >= totalGrp) return;  // wave-uniform: EXEC stays all-ones for WMMA

  const int gm = (int)(grp / ngrpN);
  const int gn = (int)(grp % ngrpN);
  const int row0 = gm << 5;
  const int col0 = gn << 6;
  const int lh = lane & 15;          // lane within half
  const int kh = (lane >> 4) << 1;   // 0 for lanes 0-15, 2 for lanes 16-31

  const float* aptr0 = A + (long)(row0 + lh) * K + kh;  // 8B-aligned pairs
  const float* aptr1 = aptr0 + (long)16 * K;
  const float* bbase = B + (long)kh * Ncols + col0 + lh;

  v8f acc[2][4] = {};
  for (int k = 0; k < K; k += 4) {
    const v2f a0 = *(const v2f*)(aptr0 + k);
    const v2f a1 = *(const v2f*)(aptr1 + k);
    const float* bk0 = bbase + (long)k * Ncols;
    const float* bk1 = bk0 + Ncols;
#pragma unroll
    for (int nj = 0; nj < 4; ++nj) {
      v2f b;
      b.x = bk0[nj * 16];
      b.y = bk1[nj * 16];
      acc[0][nj] = __builtin_amdgcn_wmma_f32_16x16x4_f32(
          false, a0, false, b, (short)0, acc[0][nj], false, false);
      acc[1][nj] = __builtin_amdgcn_wmma_f32_16x16x4_f32(
          false, a1, false, b, (short)0, acc[1][nj], false, false);
    }
  }

  const int subRow = (lane >> 4) << 3;  // 0 or 8
#pragma unroll
  for (int mi = 0; mi < 2; ++mi) {
    float* dbase = D + (long)(row0 + mi * 16 + subRow) * Ncols + col0 + lh;
#pragma unroll
    for (int nj = 0; nj < 4; ++nj) {
#pragma unroll
      for (int v = 0; v < 8; ++v)
        dbase[(long)v * Ncols + nj * 16] = acc[mi][nj][v];
    }
  }
}

// ---------------- aggregation ----------------
// init: out[n,c] = bias[c] + dis[n]^2 * h[n,c]   (bias + self-loop fused)

template <int C>
__global__ void agg_init_kernel(const float* __restrict__ h, const float* __restrict__ dis,
                                const float* __restrict__ bias, float* __restrict__ out, int n) {
  const long idx = (long)blockIdx.x * blockDim.x + threadIdx.x;  // one float4
  const long total = (long)n * (C / 4);
  if (idx >= total) return;
  const int node = (int)(idx / (C / 4));
  const int c4 = (int)(idx % (C / 4)) * 4;
  float w = dis[node];
  w = w * w;
  const float4 hv = *(const float4*)(h + (long)node * C + c4);
  const float4 bv = *(const float4*)(bias + c4);
  float4 o;
  o.x = bv.x + w * hv.x;
  o.y = bv.y + w * hv.y;
  o.z = bv.z + w * hv.z;
  o.w = bv.w + w * hv.w;
  *(float4*)(out + (long)node * C + c4) = o;
}

// edges: one wave per edge, 32 lanes cover C columns with float4 loads,
// scatter with fp32 global atomics (output buffer fits in 192MB L2).

template <int C>
__global__ void agg_edges_kernel(const float* __restrict__ h, const int* __restrict__ src,
                                 const int* __restrict__ dst, const float* __restrict__ dis,
                                 float* __restrict__ out, int E) {
  const int lane = threadIdx.x & 31;
  const long e = (long)blockIdx.x * (blockDim.x >> 5) + (threadIdx.x >> 5);
  if (e >= E) return;  // wave-uniform
  const int s = src[e];
  const int d = dst[e];
  const float coef = dis[s] * dis[d];
  const float* hs = h + (long)s * C;
  float* od = out + (long)d * C;
  if constexpr (C >= 128) {
#pragma unroll
    for (int i = 0; i < C / 128; ++i) {
      const int c = i * 128 + lane * 4;
      const float4 hv = *(const float4*)(hs + c);
      atomicAdd(od + c + 0, coef * hv.x);
      atomicAdd(od + c + 1, coef * hv.y);
      atomicAdd(od + c + 2, coef * hv.z);
      atomicAdd(od + c + 3, coef * hv.w);
    }
  } else {  // C == 64: 2 floats per lane
    const int c = lane * 2;
    const float2 hv = *(const float2*)(hs + c);
    atomicAdd(od + c + 0, coef * hv.x);
    atomicAdd(od + c + 1, coef * hv.y);
  }
}

__global__ void relu_kernel(float* __restrict__ buf, long n4) {
  const long idx = (long)blockIdx.x * blockDim.x + threadIdx.x;
  if (idx >= n4) return;
  float4 v = *(const float4*)(buf + idx * 4);
  v.x = fmaxf(v.x, 0.0f);
  v.y = fmaxf(v.y, 0.0f);
  v.z = fmaxf(v.z, 0.0f);
  v.w = fmaxf(v.w, 0.0f);
  *(float4*)(buf + idx * 4) = v;
}

// ---------------------------------------------------------------------------

extern "C" void kernel_launch(void* const* d_in, const int* in_sizes, int n_in,
                              void* d_out, int out_size, void* d_ws, size_t ws_size,
                              hipStream_t stream) {
  const float* x  = (const float*)d_in[0];
  const int*   ei = (const int*)d_in[1];
  const float* W1 = (const float*)d_in[2];
  const float* b1 = (const float*)d_in[3];
  const float* W2 = (const float*)d_in[4];
  const float* b2 = (const float*)d_in[5];
  const float* W3 = (const float*)d_in[6];
  const float* b3 = (const float*)d_in[7];
  float* out = (float*)d_out;

  const int N = GCN_N_NODES;
  const int E = in_sizes[1] / 2;
  const int* src = ei;
  const int* dst = ei + E;

  // Workspace layout: dis[N] | bufA[N*256] | bufB[N*256]
  char* ws = (char*)d_ws;
  float* dis = (float*)ws;
  size_t off = (((size_t)N * sizeof(float)) + 255) & ~(size_t)255;
  const size_t bigsz = (size_t)N * 256 * sizeof(float);
  float* bufA = (float*)(ws + off);
  float* bufB = (float*)(ws + off + bigsz);

  const int T = 256;

  // --- normalization coefficients: dis[n] = rsqrt(1 + in_degree(n)) ---
  deg_init_kernel<<<(N + T - 1) / T, T, 0, stream>>>(dis, N);
  deg_edges_kernel<<<(E + T - 1) / T, T, 0, stream>>>(dst, dis, E);
  deg_rsqrt_kernel<<<(N + T - 1) / T, T, 0, stream>>>(dis, N);

  auto launch_gemm = [&](const float* A, const float* B, float* D, int Ncols, int K) {
    const long grps = (long)(N / 32) * (Ncols / 64);
    const int blocks = (int)((grps + 7) / 8);  // 8 waves per 256-thread block
    gemm_wmma_f32<<<blocks, T, 0, stream>>>(A, B, D, N, Ncols, K);
  };

  const int edgeBlocks = (E + 7) / 8;  // 8 waves (edges) per block

  // --- layer 1: in 128 -> out 256, ReLU ---
  launch_gemm(x, W1, bufA, 256, 128);
  agg_init_kernel<256><<<(N * (256 / 4) + T - 1) / T, T, 0, stream>>>(bufA, dis, b1, bufB, N);
  agg_edges_kernel<256><<<edgeBlocks, T, 0, stream>>>(bufA, src, dst, dis, bufB, E);
  relu_kernel<<<(N * (256 / 4) + T - 1) / T, T, 0, stream>>>(bufB, (long)N * (256 / 4));

  // --- layer 2: in 256 -> out 128, ReLU ---
  launch_gemm(bufB, W2, bufA, 128, 256);
  agg_init_kernel<128><<<(N * (128 / 4) + T - 1) / T, T, 0, stream>>>(bufA, dis, b2, bufB, N);
  agg_edges_kernel<128><<<edgeBlocks, T, 0, stream>>>(bufA, src, dst, dis, bufB, E);
  relu_kernel<<<(N * (128 / 4) + T - 1) / T, T, 0, stream>>>(bufB, (long)N * (128 / 4));

  // --- layer 3: in 128 -> out 64, no activation, straight into d_out ---
  launch_gemm(bufB, W3, bufA, 64, 128);
  agg_init_kernel<64><<<(N * (64 / 4) + T - 1) / T, T, 0, stream>>>(bufA, dis, b3, out, N);
  agg_edges_kernel<64><<<edgeBlocks, T, 0, stream>>>(bufA, src, dst, dis, out, E);

  (void)n_in; (void)out_size; (void)ws_size;
}